// RGCNLayer_24893630448146
// MI455X (gfx1250) — compile-verified
//
#include <hip/hip_runtime.h>

#define N_NODES 50000
#define N_EDGES 1600000
#define IN_FEAT 128
#define OUT_FEAT 128
#define NUM_RELS 8

typedef __attribute__((ext_vector_type(2))) float v2f;
typedef __attribute__((ext_vector_type(8))) float v8f;

// Guaranteed-native f32 scatter-add: CDNA5 GLOBAL_ATOMIC_ADD_F32, no-return
// form (tracked on STOREcnt), device scope so the RMW resolves in the global
// L2 atomic units across all WGPs. s_endpgm's implicit wait-idle fences it
// before the next kernel.
__device__ __forceinline__ void atomic_add_f32_dev(float* p, float v) {
  asm volatile("global_atomic_add_f32 %0, %1, off scope:SCOPE_DEV"
               :
               : "v"((unsigned long long)(uintptr_t)p), "v"(v)
               : "memory");
}

// ---------------------------------------------------------------------------
// Kernel 1: H[r] = h @ W[r] for all relations, f32 WMMA 16x16x4.
// grid = (N_NODES/16, NUM_RELS), block = 256 (8 waves).
// Each block: one 16-node tile, one relation; wave w computes out-feature
// columns [16w, 16w+16). h tile staged in LDS (stride 132 -> bank-conflict
// free ds_load_2addr_b64 A-operand reads, 16B-aligned float4 fills).
// ---------------------------------------------------------------------------
#define LDS_STRIDE 132

__global__ __launch_bounds__(256) void rgcn_gemm_wmma(
    const float* __restrict__ h, const float* __restrict__ W,
    float* __restrict__ H) {
  __shared__ float sh[16 * LDS_STRIDE];

  const int node_base = blockIdx.x * 16;
  const int r = blockIdx.y;
  const int t = threadIdx.x;
  const int wave = t >> 5;       // 0..7 -> out-feature tile
  const int lane = t & 31;
  const int row16 = lane & 15;   // M within tile
  const int hi = lane >> 4;      // K sub-phase per WMMA f32 A/B layout
  const int n0 = wave * 16;      // out-feature base for this wave

  // Cooperative, coalesced fill of the 16x128 h tile (2 float4 per thread).
  {
    const int row = t >> 5;          // 0..7
    const int col = (t & 31) * 4;    // 0..124
#pragma unroll
    for (int it = 0; it < 2; ++it) {
      const int rr = row + it * 8;
      const float4 v =
          *(const float4*)(h + (size_t)(node_base + rr) * IN_FEAT + col);
      *(float4*)&sh[rr * LDS_STRIDE + col] = v;
    }
  }
  __syncthreads();

  const float* Wr = W + (size_t)r * IN_FEAT * OUT_FEAT;

  v8f c = {};
#pragma unroll 8
  for (int k0 = 0; k0 < IN_FEAT; k0 += 4) {
    const int ka = k0 + 2 * hi;                 // this lane's K pair base
    // A-operand: lane holds h[row16][ka], h[row16][ka+1]  (ds_load_b64)
    const v2f a = *(const v2f*)&sh[row16 * LDS_STRIDE + ka];
    // B-operand: lane holds W[ka][n0+row16], W[ka+1][n0+row16]
    v2f b;
    b.x = Wr[(size_t)ka * OUT_FEAT + n0 + row16];
    b.y = Wr[(size_t)(ka + 1) * OUT_FEAT + n0 + row16];
    // D = A*B + C  (v_wmma_f32_16x16x4_f32)
    c = __builtin_amdgcn_wmma_f32_16x16x4_f32(
        /*neg_a=*/false, a, /*neg_b=*/false, b,
        /*c_mod=*/(short)0, c, /*reuse_a=*/false, /*reuse_b=*/false);
  }

  // D layout: VGPR v -> row (v + 8*hi), col row16. Coalesced 64B per half-wave.
  float* Hout =
      H + ((size_t)r * N_NODES + node_base) * OUT_FEAT + n0 + row16;
#pragma unroll
  for (int v = 0; v < 8; ++v) {
    Hout[(size_t)(v + 8 * hi) * OUT_FEAT] = c[v];
  }
}

// ---------------------------------------------------------------------------
// Kernel 2: zero the output accumulator (float4-wide).
// ---------------------------------------------------------------------------
__global__ void rgcn_zero(float4* __restrict__ out, int n4) {
  const int i = blockIdx.x * blockDim.x + threadIdx.x;
  if (i < n4) out[i] = make_float4(0.f, 0.f, 0.f, 0.f);
}

// ---------------------------------------------------------------------------
// Kernel 3: per-edge gather + bias + norm, scatter-add via native f32 atomics.
// One wave32 per edge; each lane owns 4 contiguous output features.
// ---------------------------------------------------------------------------
__global__ __launch_bounds__(256) void rgcn_edge_scatter(
    const float* __restrict__ H, const int* __restrict__ src,
    const int* __restrict__ dst, const int* __restrict__ etype,
    const float* __restrict__ norm, const float* __restrict__ bias,
    float* __restrict__ out, int n_edges) {
  const int wave = threadIdx.x >> 5;
  const int lane = threadIdx.x & 31;
  const int e = blockIdx.x * (blockDim.x >> 5) + wave;
  if (e >= n_edges) return;

  const int s = src[e];
  const int d = dst[e];
  const int r = etype[e];
  const float nrm = norm[e];

  const float4 m =
      ((const float4*)(H + ((size_t)r * N_NODES + s) * OUT_FEAT))[lane];
  const float4 bb = ((const float4*)(bias + (size_t)r * OUT_FEAT))[lane];

  float* o = out + (size_t)d * OUT_FEAT + lane * 4;
  atomic_add_f32_dev(o + 0, (m.x + bb.x) * nrm);
  atomic_add_f32_dev(o + 1, (m.y + bb.y) * nrm);
  atomic_add_f32_dev(o + 2, (m.z + bb.z) * nrm);
  atomic_add_f32_dev(o + 3, (m.w + bb.w) * nrm);
}

// ---------------------------------------------------------------------------
// Kernel 4: ReLU epilogue (float4-wide).
// ---------------------------------------------------------------------------
__global__ void rgcn_relu(float4* __restrict__ out, int n4) {
  const int i = blockIdx.x * blockDim.x + threadIdx.x;
  if (i < n4) {
    float4 v = out[i];
    v.x = fmaxf(v.x, 0.f);
    v.y = fmaxf(v.y, 0.f);
    v.z = fmaxf(v.z, 0.f);
    v.w = fmaxf(v.w, 0.f);
    out[i] = v;
  }
}

extern "C" void kernel_launch(void* const* d_in, const int* in_sizes, int n_in,
                              void* d_out, int out_size, void* d_ws,
                              size_t ws_size, hipStream_t stream) {
  const float* h = (const float*)d_in[0];
  const int* src = (const int*)d_in[1];
  const int* dst = (const int*)d_in[2];
  const int* etype = (const int*)d_in[3];
  const float* norm = (const float*)d_in[4];
  const float* W = (const float*)d_in[5];
  const float* bias = (const float*)d_in[6];
  float* out = (float*)d_out;

  // H intermediate: [R, N, OUT] f32 = 204.8 MB in workspace.
  float* H = (float*)d_ws;

  // 1) Per-relation GEMMs via WMMA.
  dim3 ggrid(N_NODES / 16, NUM_RELS);
  rgcn_gemm_wmma<<<ggrid, 256, 0, stream>>>(h, W, H);

  // 2) Zero accumulator.
  const int n4 = (N_NODES * OUT_FEAT) / 4;
  rgcn_zero<<<(n4 + 255) / 256, 256, 0, stream>>>((float4*)out, n4);

  // 3) Edge gather/scatter, one wave per edge (8 edges per 256-thread block).
  const int edges_per_block = 256 / 32;
  rgcn_edge_scatter<<<(N_EDGES + edges_per_block - 1) / edges_per_block, 256,
                      0, stream>>>(H, src, dst, etype, norm, bias, out,
                                   N_EDGES);

  // 4) ReLU.
  rgcn_relu<<<(n4 + 255) / 256, 256, 0, stream>>>((float4*)out, n4);
}